// MambaClassifier_1125281431987
// MI455X (gfx1250) — compile-verified
//
#include <hip/hip_runtime.h>
#include <hip/hip_bf16.h>
#include <math.h>

// ---------------------------------------------------------------------------
// Mamba classifier, MI455X (gfx1250, wave32).
//   B=8 L=512 D_IN=128 D_MODEL=512 N_LAYERS=2 D_INNER=1024 D_STATE=16
//   D_CONV=4 DT_RANK=32  => M = B*L = 4096 rows for every GEMM.
// GEMMs: V_WMMA_F32_16X16X4_F32 (full fp32 — workload is scan-latency bound,
// not matmul bound, so fp32 WMMA is free) with LDS-staged double-buffered
// tiles. Staging uses CDNA5 async global->LDS loads (ASYNCcnt).
// ---------------------------------------------------------------------------

#define B_ 8
#define L_ 512
#define D_IN 128
#define D_MODEL 512
#define D_INNER 1024
#define D_STATE 16
#define D_CONV 4
#define DT_RANK 32
#define M_ROWS (B_ * L_)   // 4096

typedef float v2f __attribute__((ext_vector_type(2)));
typedef float v8f __attribute__((ext_vector_type(8)));
typedef int   v4i __attribute__((ext_vector_type(4)));

#if defined(__AMDGCN__) && __has_builtin(__builtin_amdgcn_wmma_f32_16x16x4_f32)
#define USE_WMMA_F32 1
#else
#define USE_WMMA_F32 0
#endif

#if defined(__AMDGCN__) && __has_builtin(__builtin_amdgcn_global_load_async_to_lds_b128)
#define HAVE_ASYNC_LDS 1
#else
#define HAVE_ASYNC_LDS 0
#endif

// 16-byte global -> LDS copy. Async path goes straight to LDS (ASYNCcnt),
// fallback stages through VGPRs (global_load_b128 + ds_store_b128).
// Builtin signature (from clang diagnostic): (v4i as1*, v4i as3*, Imm, Imm).
__device__ __forceinline__ void copy16_to_lds(const float* g, float* l) {
#if HAVE_ASYNC_LDS
  __builtin_amdgcn_global_load_async_to_lds_b128(
      (__attribute__((address_space(1))) v4i*)g,
      (__attribute__((address_space(3))) v4i*)l, 0, 0);
#else
  *(float4*)l = *(const float4*)g;
#endif
}

__device__ __forceinline__ void staging_wait() {
#if HAVE_ASYNC_LDS
#if __has_builtin(__builtin_amdgcn_s_wait_asynccnt)
  __builtin_amdgcn_s_wait_asynccnt(0);
#else
  asm volatile("s_wait_asynccnt 0x0" ::: "memory");
#endif
#endif
}

// ---------------------------------------------------------------------------
// fp32 WMMA GEMM:  C[M x N] = act( A[M x K] * W[N x K]^T + bias )
// Block = 128 threads = 4 waves. Block tile 64(M) x 64(N); wave w owns
// 16(M) x 64(N) = four 16x16 WMMA accumulators. K is consumed in BK=32
// slabs staged into double-buffered LDS (stride padded to 36 floats: keeps
// 16B alignment for b128 staging and makes the 16 fragment rows hit 16
// distinct banks for conflict-free ds_load_b64).
// ACT: 0 = none, 1 = softplus (dt_proj).
// ---------------------------------------------------------------------------
template <int ACT, int HASB>
__global__ __launch_bounds__(128) void gemm_f32_wmma(
    const float* __restrict__ A, const float* __restrict__ W,
    const float* __restrict__ bias, float* __restrict__ C,
    int M, int N, int K, int lda, int ldc) {
  (void)M; (void)N;
  constexpr int BK = 32;
  constexpr int STRIDE = 36;                 // padded LDS row stride (floats)
  __shared__ float sA[2][64 * STRIDE];
  __shared__ float sB[2][64 * STRIDE];

  const int tid  = threadIdx.x;
  const int lane = tid & 31;
  const int wave = tid >> 5;
  const int row  = lane & 15;                // M index (A frag) / N index (B frag)
  const int hi   = lane >> 4;                // K-pair select per ISA 16x4 layout
  const int koff = 2 * hi;

  const int mblk = blockIdx.y * 64;
  const int nblk = blockIdx.x * 64;

  // Cooperative staging: tile = 64 rows x 32 floats = 512 float4 chunks;
  // 128 threads x 4 chunks, fully coalesced 16B per lane.
  auto stage = [&](int kt, int buf) {
    const int kb = kt * BK;
#pragma unroll
    for (int c = 0; c < 4; ++c) {
      const int id = c * 128 + tid;
      const int r  = id >> 3;
      const int cx = (id & 7) * 4;
      const float* ga = A + (size_t)(mblk + r) * lda + kb + cx;
      const float* gw = W + (size_t)(nblk + r) * K + kb + cx;
      copy16_to_lds(ga, &sA[buf][r * STRIDE + cx]);
      copy16_to_lds(gw, &sB[buf][r * STRIDE + cx]);
      __builtin_prefetch(ga + BK, 0, 1);     // touch tile kt+2 (global_prefetch_b8)
    }
  };

  v8f acc[4] = {v8f{}, v8f{}, v8f{}, v8f{}};
  const int nk = K / BK;

  stage(0, 0);
  int buf = 0;
  for (int kt = 0; kt < nk; ++kt) {
    staging_wait();          // this wave's async loads have landed in LDS
    __syncthreads();         // all waves' staging done; prev-tile reads done
    if (kt + 1 < nk) stage(kt + 1, buf ^ 1);   // overlap next slab with math

#if USE_WMMA_F32
    const float* a_base = &sA[buf][(wave * 16 + row) * STRIDE + koff];
    const float* b_base = &sB[buf][row * STRIDE + koff];
#pragma unroll
    for (int k0 = 0; k0 < BK; k0 += 4) {
      v2f a  = *(const v2f*)(a_base + k0);
      v2f b0 = *(const v2f*)(b_base + k0);
      v2f b1 = *(const v2f*)(b_base + 16 * STRIDE + k0);
      v2f b2 = *(const v2f*)(b_base + 32 * STRIDE + k0);
      v2f b3 = *(const v2f*)(b_base + 48 * STRIDE + k0);
      acc[0] = __builtin_amdgcn_wmma_f32_16x16x4_f32(false, a, false, b0, (short)0, acc[0], false, false);
      acc[1] = __builtin_amdgcn_wmma_f32_16x16x4_f32(false, a, false, b1, (short)0, acc[1], false, false);
      acc[2] = __builtin_amdgcn_wmma_f32_16x16x4_f32(false, a, false, b2, (short)0, acc[2], false, false);
      acc[3] = __builtin_amdgcn_wmma_f32_16x16x4_f32(false, a, false, b3, (short)0, acc[3], false, false);
    }
#else
    // Scalar fallback (host pass / builtin unavailable): lane owns
    // C[mblk+wave*16+v+8*hi][nblk+nt*16+row] for v in 0..7.
#pragma unroll
    for (int k0 = 0; k0 < BK; k0 += 4) {
#pragma unroll
      for (int kk = 0; kk < 4; ++kk) {
#pragma unroll
        for (int nt = 0; nt < 4; ++nt) {
          const float bv = sB[buf][(nt * 16 + row) * STRIDE + k0 + kk];
#pragma unroll
          for (int v = 0; v < 8; ++v)
            acc[nt][v] += sA[buf][(wave * 16 + v + 8 * hi) * STRIDE + k0 + kk] * bv;
        }
      }
    }
#endif
    buf ^= 1;
  }

  // Epilogue: C/D layout — VGPR v holds M=v (lanes 0-15) / M=v+8 (lanes 16-31)
#pragma unroll
  for (int nt = 0; nt < 4; ++nt) {
    const int col = nblk + nt * 16 + row;
    const float bcol = HASB ? bias[col] : 0.0f;
#pragma unroll
    for (int v = 0; v < 8; ++v) {
      const int r = mblk + wave * 16 + v + 8 * hi;
      float val = acc[nt][v] + bcol;
      if (ACT == 1) val = (val > 20.0f) ? val : log1pf(__expf(val));  // softplus
      C[(size_t)r * ldc + col] = val;
    }
  }
}

// ---------------------------------------------------------------------------
// Depthwise causal conv1d (kernel 4, pad-left 3) + SiLU.
// xz layout (b,l,2048): xi = [:,:,0:1024]. Output xc (b,l,1024).
// One thread per (b,l,d); d fastest => fully coalesced.
// ---------------------------------------------------------------------------
__global__ __launch_bounds__(256) void conv_silu_kernel(
    const float* __restrict__ xz, const float* __restrict__ cw,
    const float* __restrict__ cb, float* __restrict__ xc) {
  const unsigned idx = blockIdx.x * 256u + threadIdx.x;
  const int d = idx & (D_INNER - 1);
  const int l = (idx >> 10) & (L_ - 1);
  const int b = idx >> 19;
  float acc = cb[d];
#pragma unroll
  for (int j = 0; j < D_CONV; ++j) {
    const int li = l + j - (D_CONV - 1);
    if (li >= 0)
      acc += cw[d * D_CONV + j] * xz[(size_t)(b * L_ + li) * (2 * D_INNER) + d];
  }
  const float sg = 1.0f / (1.0f + __expf(-acc));
  xc[(size_t)(b * L_ + l) * D_INNER + d] = acc * sg;
}

// ---------------------------------------------------------------------------
// Selective scan, fused with skip (D*x) and SiLU(z) gate.
// One thread per (b, d): 16 states in VGPRs, 512 sequential steps.
// Per-step B_t/C_t (32 floats, common to the whole block) staged via LDS.
// Grid = (D_INNER/256, B), block = 256 (8 waves, wave32).
// ---------------------------------------------------------------------------
__global__ __launch_bounds__(256) void scan_kernel(
    const float* __restrict__ dtb,  // softplus(dt) (b,l,1024)
    const float* __restrict__ xcb,  // conv+silu output (b,l,1024)
    const float* __restrict__ xzb,  // gate z lives at (b,l,1024 + d)
    const float* __restrict__ xdbl, // (b,l,64): [0:32)=dt_low [32:48)=B [48:64)=C
    const float* __restrict__ A_log, const float* __restrict__ Dm,
    float* __restrict__ yb) {
  const int d = blockIdx.x * 256 + threadIdx.x;
  const int b = blockIdx.y;

  float Arow[D_STATE];
#pragma unroll
  for (int s = 0; s < D_STATE; ++s) Arow[s] = -__expf(A_log[d * D_STATE + s]);
  const float Dd = Dm[d];

  float st[D_STATE];
#pragma unroll
  for (int s = 0; s < D_STATE; ++s) st[s] = 0.0f;

  __shared__ float sBC[32];  // [0:16)=B_t, [16:32)=C_t

  for (int l = 0; l < L_; ++l) {
    __syncthreads();
    if (threadIdx.x < 32)
      sBC[threadIdx.x] = xdbl[(size_t)(b * L_ + l) * 64 + DT_RANK + threadIdx.x];
    __syncthreads();

    const size_t r = (size_t)(b * L_ + l);
    const float dt = dtb[r * D_INNER + d];
    const float x  = xcb[r * D_INNER + d];
    const float z  = xzb[r * (2 * D_INNER) + D_INNER + d];
    const float dtx = dt * x;
    float yv = 0.0f;
#pragma unroll
    for (int s = 0; s < D_STATE; ++s) {
      st[s] = __expf(dt * Arow[s]) * st[s] + dtx * sBC[s];
      yv += st[s] * sBC[D_STATE + s];
    }
    const float gate = z * (1.0f / (1.0f + __expf(-z)));   // silu(z)
    yb[r * D_INNER + d] = (yv + x * Dd) * gate;
  }
}

// Mean over L: (B,L,D_MODEL) -> (B,D_MODEL)
__global__ __launch_bounds__(256) void mean_kernel(
    const float* __restrict__ h, float* __restrict__ hm) {
  const int idx = blockIdx.x * 256 + threadIdx.x;  // 4096 threads
  const int b = idx >> 9, dm = idx & (D_MODEL - 1);
  float s = 0.0f;
  for (int l = 0; l < L_; ++l) s += h[(size_t)(b * L_ + l) * D_MODEL + dm];
  hm[b * D_MODEL + dm] = s * (1.0f / (float)L_);
}

// Classifier head: Linear(512->64) + ReLU + Linear(64->1). One block per batch.
__global__ __launch_bounds__(64) void head_kernel(
    const float* __restrict__ hm, const float* __restrict__ w1,
    const float* __restrict__ b1, const float* __restrict__ w2,
    const float* __restrict__ b2, float* __restrict__ out) {
  const int b = blockIdx.x, j = threadIdx.x;
  float acc = b1[j];
  for (int k = 0; k < D_MODEL; ++k) acc += hm[b * D_MODEL + k] * w1[j * D_MODEL + k];
  __shared__ float sh[64];
  sh[j] = fmaxf(acc, 0.0f) * w2[j];
  __syncthreads();
  if (j == 0) {
    float s = b2[0];
    for (int t = 0; t < 64; ++t) s += sh[t];
    out[b] = s;
  }
}

// ---------------------------------------------------------------------------
// Host side
// ---------------------------------------------------------------------------
extern "C" void kernel_launch(void* const* d_in, const int* in_sizes, int n_in,
                              void* d_out, int out_size, void* d_ws, size_t ws_size,
                              hipStream_t stream) {
  (void)in_sizes; (void)n_in; (void)out_size; (void)ws_size;
  const float* x    = (const float*)d_in[0];
  const float* piw  = (const float*)d_in[1];
  const float* pib  = (const float*)d_in[2];
  const float* inw  = (const float*)d_in[3];
  const float* cw   = (const float*)d_in[4];
  const float* cb   = (const float*)d_in[5];
  const float* xpw  = (const float*)d_in[6];
  const float* dtw  = (const float*)d_in[7];
  const float* dtbi = (const float*)d_in[8];
  const float* alog = (const float*)d_in[9];
  const float* dmm  = (const float*)d_in[10];
  const float* outw = (const float*)d_in[11];
  const float* w1   = (const float*)d_in[12];
  const float* b1   = (const float*)d_in[13];
  const float* w2   = (const float*)d_in[14];
  const float* b2   = (const float*)d_in[15];

  // Workspace layout (bytes). Total ~93.3 MB — L2-resident on MI455X (192 MB).
  char* ws = (char*)d_ws;
  float* h    = (float*)(ws);                                    // 4096*512
  float* xz   = (float*)(ws + (size_t)8  * 1024 * 1024);         // 4096*2048
  float* xc   = (float*)(ws + (size_t)42 * 1024 * 1024);         // 4096*1024
  float* xdbl = (float*)(ws + (size_t)59 * 1024 * 1024);         // 4096*64
  float* dtb  = (float*)(ws + (size_t)61 * 1024 * 1024);         // 4096*1024
  float* yb   = (float*)(ws + (size_t)78 * 1024 * 1024);         // 4096*1024
  float* hm   = (float*)(ws + (size_t)95 * 1024 * 1024);         // 8*512

  const dim3 blk(128);

  // h = x @ proj_in_w^T + b          (M=4096, N=512, K=128)
  gemm_f32_wmma<0, 1><<<dim3(D_MODEL / 64, M_ROWS / 64), blk, 0, stream>>>(
      x, piw, pib, h, M_ROWS, D_MODEL, D_IN, D_IN, D_MODEL);

  for (int l = 0; l < 2; ++l) {
    const float* inw_l  = inw  + (size_t)l * 2 * D_INNER * D_MODEL;
    const float* cw_l   = cw   + (size_t)l * D_INNER * D_CONV;
    const float* cb_l   = cb   + (size_t)l * D_INNER;
    const float* xpw_l  = xpw  + (size_t)l * (DT_RANK + 2 * D_STATE) * D_INNER;
    const float* dtw_l  = dtw  + (size_t)l * D_INNER * DT_RANK;
    const float* dtb_l  = dtbi + (size_t)l * D_INNER;
    const float* alog_l = alog + (size_t)l * D_INNER * D_STATE;
    const float* dmm_l  = dmm  + (size_t)l * D_INNER;
    const float* outw_l = outw + (size_t)l * D_MODEL * D_INNER;

    // xz = h @ in_proj^T               (N=2048, K=512)
    gemm_f32_wmma<0, 0><<<dim3(2 * D_INNER / 64, M_ROWS / 64), blk, 0, stream>>>(
        h, inw_l, nullptr, xz, M_ROWS, 2 * D_INNER, D_MODEL, D_MODEL, 2 * D_INNER);

    // xc = silu(causal depthwise conv(xi) + cb)
    conv_silu_kernel<<<(B_ * L_ * D_INNER) / 256, 256, 0, stream>>>(xz, cw_l, cb_l, xc);

    // x_dbl = xc @ x_proj^T            (N=64, K=1024)
    gemm_f32_wmma<0, 0><<<dim3(1, M_ROWS / 64), blk, 0, stream>>>(
        xc, xpw_l, nullptr, xdbl, M_ROWS, 64, D_INNER, D_INNER, 64);

    // dt = softplus(dt_low @ dt_proj^T + dt_b)   (N=1024, K=32, lda=64)
    gemm_f32_wmma<1, 1><<<dim3(D_INNER / 64, M_ROWS / 64), blk, 0, stream>>>(
        xdbl, dtw_l, dtb_l, dtb, M_ROWS, D_INNER, DT_RANK, 64, D_INNER);

    // selective scan + D-skip + silu(z) gate
    scan_kernel<<<dim3(D_INNER / 256, B_), 256, 0, stream>>>(
        dtb, xc, xz, xdbl, alog_l, dmm_l, yb);

    // h = y @ out_proj^T               (N=512, K=1024)
    gemm_f32_wmma<0, 0><<<dim3(D_MODEL / 64, M_ROWS / 64), blk, 0, stream>>>(
        yb, outw_l, nullptr, h, M_ROWS, D_MODEL, D_INNER, D_INNER, D_MODEL);
  }

  mean_kernel<<<(B_ * D_MODEL) / 256, 256, 0, stream>>>(h, hm);
  head_kernel<<<B_, 64, 0, stream>>>(hm, w1, b1, w2, b2, (float*)d_out);
}